// BilinearInterpolation_26603027432054
// MI455X (gfx1250) — compile-verified
//
#include <hip/hip_runtime.h>
#include <cstdint>

// Fused bilinear demosaic, one pass.
//  - Interior tiles: stage-1 halo tile (72x20 fp32, row stride 1024) is DMA'd
//    global->LDS by ONE Tensor-Data-Mover instruction (tensor_load_to_lds,
//    TENSORcnt) issued by wave 0.
//  - Border tiles: zero-fill ('SAME' padding) + predicated per-lane
//    global_load_async_to_lds_b128 (ASYNCcnt).
//  - Stencils run out of LDS; mask/out are nontemporal 128b streaming ops.
// HBM-bound: ~604 MB moved => ~26 us floor at 23.3 TB/s.

typedef float v4f __attribute__((ext_vector_type(4)));
typedef unsigned int u32x4 __attribute__((ext_vector_type(4)));
typedef unsigned int u32x8 __attribute__((ext_vector_type(8)));

#define TX 64
#define TY 16
#define SXA 72          // stage-1 tile width: halo 4 each side in x (16B rows)
#define SY  20          // stage-1 tile height: halo 2 each side in y
#define QX  (SXA / 4)   // 18 quads per s_x row
#define NQ  (QX * SY)   // 360 quads
#define UXA 68          // t tile width  (covers x = -2..65)
#define UY  18          // t tile height (covers y = -1..16)
#define QTX (UXA / 4)   // 17 quads per t row
#define NQT (QTX * UY)  // 306 quads
#define NTHREADS 256
#define IMG_W 1024
#define IMG_H 1024

// Low 32 bits of a flat shared-aperture address are the LDS byte address.
__device__ __forceinline__ unsigned lds_addr32(const void* p) {
  return (unsigned)(uintptr_t)p;
}

// CDNA5 async copy: 16 bytes global -> LDS per lane, tracked by ASYNCcnt.
__device__ __forceinline__ void async_load_b128(unsigned lds_byte_addr,
                                                const float* gptr) {
  asm volatile("global_load_async_to_lds_b128 %0, %1, off"
               :
               : "v"(lds_byte_addr), "v"((uint64_t)(uintptr_t)gptr)
               : "memory");
}

__device__ __forceinline__ void wait_asynccnt0() {
  asm volatile("s_wait_asynccnt 0" ::: "memory");
}

// CDNA5 Tensor Data Mover: one 2D tile DMA global -> LDS (TENSORcnt).
// Two-group D# form (tensors up to 2D). Issued once per wave that executes it.
__device__ __forceinline__ void tdm_load_tile_2d(unsigned lds_byte_addr,
                                                 const float* tile_gptr) {
  const uint64_t ga = (uint64_t)(uintptr_t)tile_gptr;

  u32x4 g0;
  g0[0] = 1u;                                     // count=1, no gather/restore
  g0[1] = lds_byte_addr;                          // lds_addr
  g0[2] = (unsigned)ga;                           // global_addr[31:0]
  g0[3] = (unsigned)(ga >> 32) | 0x80000000u;     // global_addr[56:32] | type=2

  u32x8 g1;
  g1[0] = 0x00020000u;                            // wg_mask=0, data_size=4B
  g1[1] = ((unsigned)IMG_W & 0xFFFFu) << 16;      // tensor_dim0[15:0]
  g1[2] = (((unsigned)IMG_W >> 16) & 0xFFFFu) |   // tensor_dim0[31:16]
          (((unsigned)IMG_H & 0xFFFFu) << 16);    // tensor_dim1[15:0]
  g1[3] = (((unsigned)IMG_H >> 16) & 0xFFFFu) |   // tensor_dim1[31:16]
          ((unsigned)SXA << 16);                  // tile_dim0 = 72
  g1[4] = (unsigned)SY;                           // tile_dim1 = 20, tile_dim2=0
  g1[5] = (unsigned)IMG_W;                        // tensor_dim0_stride[31:0]
  g1[6] = ((unsigned)((IMG_W * IMG_H) & 0xFFFF)) << 16;  // d0s[47:32]=0 | d1s[15:0]
  g1[7] = (unsigned)((unsigned long long)((unsigned long long)IMG_W * IMG_H) >> 16); // d1s[47:16]

  asm volatile("tensor_load_to_lds %0, %1"
               :
               : "s"(g0), "s"(g1)
               : "memory");
  asm volatile("s_wait_tensorcnt 0" ::: "memory");
}

__global__ __launch_bounds__(NTHREADS)
void demosaic_kernel(const float* __restrict__ mosic,
                     const float* __restrict__ mask,
                     float* __restrict__ out)
{
  __shared__ __align__(16) float s_x[SY][SXA];  // raw channel tile
  __shared__ __align__(16) float s_t[UY][UXA];  // t = x (+ diag conv)

  const int  tid  = threadIdx.x;
  const int  bc   = blockIdx.z;          // b*3 + c
  const bool isRB = ((bc % 3) != 1);
  const int  W = IMG_W, H = IMG_H;
  const int  oxb = blockIdx.x * TX;
  const int  oyb = blockIdx.y * TY;
  const int  x0  = oxb - 4;              // s_x col 0 -> global x0 (16B aligned)
  const int  y0  = oyb - 2;              // s_x row 0 -> global y0
  const size_t plane = (size_t)bc * (size_t)(H * W);
  const float* __restrict__ src = mosic + plane;

  const bool interior = (x0 >= 0) && (x0 + SXA <= W) &&
                        (y0 >= 0) && (y0 + SY  <= H);

  const unsigned sbase = lds_addr32(&s_x[0][0]);

  if (interior) {
    // --- hot path: one TDM descriptor moves the whole tile ---
    // Scalar branch on wave id: TDM ignores EXEC, so an exec-mask guard
    // would issue 8 duplicate DMAs.
    int wave = __builtin_amdgcn_readfirstlane(tid >> 5);
    if (wave == 0)
      tdm_load_tile_2d(sbase, src + (size_t)y0 * W + x0);
    __syncthreads();   // TENSORcnt drained by wave 0 before signaling
  } else {
    // --- border path: zero padding + predicated async copies ---
    v4f* s4 = (v4f*)&s_x[0][0];
    for (int q = tid; q < NQ; q += NTHREADS) s4[q] = (v4f)0.0f;
    __syncthreads();
    for (int q = tid; q < NQ; q += NTHREADS) {
      int jy = q / QX;
      int jx = q - jy * QX;
      int gy = y0 + jy;
      int gx = x0 + jx * 4;  // quad-aligned; W%4==0 -> quad fully in or out
      if ((unsigned)gy < (unsigned)H && (unsigned)gx < (unsigned)W)
        async_load_b128(sbase + (unsigned)q * 16u, src + (size_t)gy * W + gx);
    }
    wait_asynccnt0();
    __syncthreads();
  }

  // --- stage 2: t = x + conv(x, K_RB) for R/B ; t = x for G ---
  // t[jy][jx] is pixel (y = oyb + jy - 1, x = oxb + jx - 2)
  for (int q = tid; q < NQT; q += NTHREADS) {
    int jy  = q / QTX;
    int jx4 = (q - jy * QTX) * 4;
    v4f v;
#pragma unroll
    for (int e = 0; e < 4; ++e) {
      int jx  = jx4 + e;
      float c = s_x[jy + 1][jx + 2];
      if (isRB) {
        float d = s_x[jy][jx + 1] + s_x[jy][jx + 3] +
                  s_x[jy + 2][jx + 1] + s_x[jy + 2][jx + 3];
        c = fmaf(0.25f, d, c);
      }
      v[e] = c;
    }
    *(v4f*)&s_t[jy][jx4] = v;   // ds_store_b128
  }
  __syncthreads();

  // --- stage 3: one float4 output quad per thread; streaming mask/out ---
  {
    int oy  = tid >> 4;          // 0..15
    int ox4 = (tid & 15) * 4;    // 0,4,...,60
    size_t g = plane + (size_t)(oyb + oy) * W + (size_t)(oxb + ox4);
    v4f m = __builtin_nontemporal_load((const v4f*)(mask + g));
    v4f res;
#pragma unroll
    for (int e = 0; e < 4; ++e) {
      int ox = ox4 + e;
      float t  = s_t[oy + 1][ox + 2];
      float cr = s_t[oy][ox + 2] + s_t[oy + 2][ox + 2] +
                 s_t[oy + 1][ox + 1] + s_t[oy + 1][ox + 3];
      float interp = fmaf(0.25f, cr, t);
      float xc = s_x[oy + 2][ox + 4];   // original mosic value
      float v  = fmaf(interp, 1.0f - m[e], xc);
      v = fminf(fmaxf(v, 0.0f), 255.0f);
      res[e] = v * (1.0f / 255.0f);
    }
    __builtin_nontemporal_store(res, (v4f*)(out + g));
  }
}

extern "C" void kernel_launch(void* const* d_in, const int* in_sizes, int n_in,
                              void* d_out, int out_size, void* d_ws, size_t ws_size,
                              hipStream_t stream) {
  const float* mosic = (const float*)d_in[0];
  const float* mask  = (const float*)d_in[1];
  float* out = (float*)d_out;

  const int H = IMG_H, W = IMG_W;
  const int B = in_sizes[0] / (3 * H * W);   // 16 for the reference shapes

  dim3 grid(W / TX, H / TY, B * 3);          // (16, 64, 48)
  demosaic_kernel<<<grid, NTHREADS, 0, stream>>>(mosic, mask, out);
}